// SpectralConv1d_7516192768051
// MI455X (gfx1250) — compile-verified
//
#include <hip/hip_runtime.h>
#include <hip/hip_bf16.h>
#include <stdint.h>

// SpectralConv1d as three dense GEMMs on the f32 WMMA pipe (V_WMMA_F32_16X16X4_F32),
// CDNA5 async-to-LDS staging, and K-pair-interleaved operand layouts so every
// WMMA fragment is a single aligned b64 load into an adjacent VGPR pair.
//   x:    (32,8,64,2048) f32 -> rows r = b*64+i
//   Bfp:  (1024 pair-rows x 128 cols) float2  forward DFT basis [cos | -sin]
//   Bip:  (64 pair-rows x 2048 cols) float2   inverse DFT basis (irfft scaling folded)
//   abuf: [b][spec_col][i]   (256 x 128 x 64) spectral coeffs, i-contiguous
//   ybuf: [b*64+o][spec_col] (16384 x 128)

#define SEQ      2048
#define MODES    64
#define NCH      64
#define MROWS    16384
#define SPECW    128
#define TWO_PI   6.283185307179586f

#define KC        32      // K-chunk staged per iteration in dft_fwd
#define BS_STRIDE 288     // 128 float2 (256 f) + 32 pad: lane halves -> disjoint banks
#define XS_STRIDE 36      // 32 + 4 pad: all 64 banks distinct for a-frag b64 reads
#define YS_STRIDE 132     // 128 + 4 pad

typedef __attribute__((ext_vector_type(2))) float v2f;
typedef __attribute__((ext_vector_type(8))) float v8f;

__device__ __forceinline__ v8f wmma_f32(v2f a, v2f b, v8f c) {
    return __builtin_amdgcn_wmma_f32_16x16x4_f32(
        false, a, false, b, (short)0, c, false, false);
}

// ---- CDNA5 async Global->LDS copy (ASYNCcnt-tracked, no VGPR data path) ----
__device__ __forceinline__ uint32_t lds_off(const void* p) {
    return (uint32_t)(uintptr_t)p;   // low 32 bits of shared-aperture flat addr
}
__device__ __forceinline__ void async_copy_b128(uint32_t lds_byte_off, const float* g) {
    asm volatile("global_load_async_to_lds_b128 %0, %1, off"
                 :: "v"(lds_byte_off), "v"((uint64_t)(uintptr_t)g)
                 : "memory");
}
__device__ __forceinline__ void wait_async_0() {
    asm volatile("s_wait_asynccnt 0x0" ::: "memory");
}

// ---------------------------------------------------------------------------
// Kernel 0: build pair-interleaved DFT bases.
//   Bfp float layout: (n>>1)*256 + 2*c + (n&1)         (pair of K-rows n, n+1)
//   Bip float layout: ((c>>1)*2048 + n)*2 + (c&1)      (pair of spec rows c, c+1)
// ---------------------------------------------------------------------------
__global__ void __launch_bounds__(256) build_basis(float* __restrict__ Bfp,
                                                   float* __restrict__ Bip) {
    int idx = blockIdx.x * blockDim.x + threadIdx.x;   // over 2048*64
    if (idx >= SEQ * MODES) return;
    int n = idx >> 6;
    int k = idx & 63;
    int m = (k * n) & (SEQ - 1);
    float w = TWO_PI * (float)m * (1.0f / (float)SEQ);
    float c, s;
    __sincosf(w, &s, &c);
    size_t fb = (size_t)(n >> 1) * 256 + (n & 1);
    Bfp[fb + 2 * k]           = c;
    Bfp[fb + 2 * (MODES + k)] = -s;
    float ar = (k == 0) ? (1.0f / (float)SEQ) : (2.0f / (float)SEQ);
    Bip[(((size_t)(k >> 1)) * SEQ + n) * 2 + (k & 1)]        = ar * c;
    Bip[(((size_t)(32 + (k >> 1))) * SEQ + n) * 2 + (k & 1)] =
        (k == 0) ? 0.0f : -(2.0f / (float)SEQ) * s;
}

// ---------------------------------------------------------------------------
// Kernel 0b: repack wr/wi into in-channel pair-interleaved copies.
// ---------------------------------------------------------------------------
__global__ void __launch_bounds__(256) repack_w(const float* __restrict__ wr,
                                                const float* __restrict__ wi,
                                                float* __restrict__ wrp,
                                                float* __restrict__ wip) {
    int idx = blockIdx.x * blockDim.x + threadIdx.x;   // over 32*64*64
    if (idx >= (NCH / 2) * NCH * MODES) return;
    int k  = idx & 63;
    int o  = (idx >> 6) & 63;
    int ip = idx >> 12;
    size_t src = ((size_t)(2 * ip) * NCH + o) * MODES + k;
    size_t dst = (((size_t)ip * NCH + o) * MODES + k) * 2;
    wrp[dst]     = wr[src];
    wrp[dst + 1] = wr[src + (size_t)NCH * MODES];
    wip[dst]     = wi[src];
    wip[dst + 1] = wi[src + (size_t)NCH * MODES];
}

// ---------------------------------------------------------------------------
// Kernel 1: spectral coeffs += X(16384x2048) @ Bf(2048x128).
// Block = 8 waves sharing one K-half; wave = row tile. Async-stage the Bfp slab
// (8x dedup) and each wave's 16xKC x tile; fragments are single b64 DS loads.
// B fragments are batch-preloaded per K-step so DS latency overlaps the WMMAs.
// ---------------------------------------------------------------------------
__global__ void __launch_bounds__(256) dft_fwd(const float* __restrict__ x,
                                               const float* __restrict__ Bfp,
                                               float* __restrict__ abuf) {
    __shared__ float bs[(KC / 2) * BS_STRIDE];    // 16 pair-rows -> 18.4 KB
    __shared__ float xs[8 * 16 * XS_STRIDE];      // 18.4 KB
    const int tid  = threadIdx.x;
    const int lane = tid & 31;
    const int wave = tid >> 5;
    const int ks   = blockIdx.x & 1;
    const int rt   = (blockIdx.x >> 1) * 8 + wave;
    const int row0 = rt * 16;
    const int half = lane >> 4;
    const int l15  = lane & 15;
    const int kbase0 = ks * (SEQ / 2);

    v8f acc[8] = {};
    float* xw_stage = &xs[wave * 16 * XS_STRIDE];
    const float* xw_read = &xs[wave * 16 * XS_STRIDE + l15 * XS_STRIDE];

    const int NCHUNK = (SEQ / 2) / KC;            // 32
    for (int cch = 0; cch < NCHUNK; ++cch) {
        const int kb = kbase0 + cch * KC;
        const int pb = kb >> 1;                   // pair-row base in Bfp
#pragma unroll
        for (int j = 0; j < 4; ++j) {             // Bfp: 16 pair-rows x 256 f
            int seg = tid + j * 256;
            int r = seg >> 6, c4 = (seg & 63) * 4;
            async_copy_b128(lds_off(&bs[r * BS_STRIDE + c4]),
                            Bfp + (size_t)(pb + r) * 256 + c4);
        }
#pragma unroll
        for (int j = 0; j < 4; ++j) {             // x tile: 16 rows x KC f
            int seg = lane + j * 32;
            int r = seg >> 3, c4 = (seg & 7) * 4;
            async_copy_b128(lds_off(&xw_stage[r * XS_STRIDE + c4]),
                            x + (size_t)(row0 + r) * SEQ + kb + c4);
        }
        wait_async_0();
        __syncthreads();

#pragma unroll 2
        for (int kc = 0; kc < KC; kc += 4) {
            const int ka = kc + 2 * half;         // A layout: lane half -> K pair
            const int pa = (kc >> 1) + half;      // pair-row in staged Bfp
            v2f a = *(const v2f*)&xw_read[ka];
            v2f bfr[8];
#pragma unroll
            for (int t = 0; t < 8; ++t)
                bfr[t] = *(const v2f*)&bs[pa * BS_STRIDE + (t * 16 + l15) * 2];
#pragma unroll
            for (int t = 0; t < 8; ++t)
                acc[t] = wmma_f32(a, bfr[t], acc[t]);
        }
        __syncthreads();
    }
#pragma unroll
    for (int t = 0; t < 8; ++t) {
        const int n0 = t * 16;
#pragma unroll
        for (int r = 0; r < 8; ++r) {
            const int rr = row0 + r + 8 * half;   // global row = b*64 + i
            const int b = rr >> 6, i = rr & 63;
            unsafeAtomicAdd(&abuf[((size_t)b * SPECW + n0 + l15) * NCH + i],
                            acc[t][r]);
        }
    }
}

// ---------------------------------------------------------------------------
// Kernel 2: per-mode complex channel mix (tiny: 0.5 GFLOP).
//   Yre = Are@wr - Aim@wi ;  Yim = Are@wi + Aim@wr
// ---------------------------------------------------------------------------
__global__ void __launch_bounds__(256) mode_mix(const float* __restrict__ abuf,
                                                const float* __restrict__ wrp,
                                                const float* __restrict__ wip,
                                                float* __restrict__ ybuf) {
    const int lane = threadIdx.x & 31;
    const int wave = threadIdx.x >> 5;
    const int job  = blockIdx.x * 8 + wave;     // 64 modes * 16 batch tiles
    const int k    = job >> 4;
    const int b0   = (job & 15) * 16;
    const int half = lane >> 4;
    const int l15  = lane & 15;

    const v2f* wr2 = (const v2f*)wrp;
    const v2f* wi2 = (const v2f*)wip;

    v8f accRe[4] = {};
    v8f accIm[4] = {};

    for (int kk = 0; kk < NCH; kk += 4) {
        const int i0 = kk + 2 * half;
        const int ip = (kk >> 1) + half;
        const size_t abase = (size_t)(b0 + l15) * SPECW;
        v2f aRe = *(const v2f*)&abuf[(abase + k) * NCH + i0];
        v2f aIm = *(const v2f*)&abuf[(abase + MODES + k) * NCH + i0];
        v2f bR[4], bI[4];
#pragma unroll
        for (int t = 0; t < 4; ++t) {
            const size_t wb = ((size_t)ip * NCH + t * 16 + l15) * MODES + k;
            bR[t] = wr2[wb];
            bI[t] = wi2[wb];
        }
#pragma unroll
        for (int t = 0; t < 4; ++t) {
            v2f bIn; bIn.x = -bI[t].x; bIn.y = -bI[t].y;
            accRe[t] = wmma_f32(aRe, bR[t], accRe[t]);
            accRe[t] = wmma_f32(aIm, bIn,   accRe[t]);
            accIm[t] = wmma_f32(aRe, bI[t], accIm[t]);
            accIm[t] = wmma_f32(aIm, bR[t], accIm[t]);
        }
    }
#pragma unroll
    for (int t = 0; t < 4; ++t) {
        const int o0 = t * 16;
#pragma unroll
        for (int r = 0; r < 8; ++r) {
            const int b = b0 + r + 8 * half;
            const size_t row = (size_t)b * NCH + o0 + l15;   // ybuf row = b*64+o
            ybuf[row * SPECW + k]         = accRe[t][r];
            ybuf[row * SPECW + MODES + k] = accIm[t][r];
        }
    }
}

// ---------------------------------------------------------------------------
// Kernel 3: out(16384x2048) = ybuf(16384x128) @ Bi(128x2048).
// Block = 8 waves sharing one 16x128 y tile (async-staged once to LDS);
// Bi fragments are coalesced global b64 loads from the pair-interleaved Bip.
// ---------------------------------------------------------------------------
__global__ void __launch_bounds__(256) dft_inv(const float* __restrict__ ybuf,
                                               const float* __restrict__ Bip,
                                               float* __restrict__ out) {
    __shared__ float ys[16 * YS_STRIDE];          // 8.4 KB
    const int tid  = threadIdx.x;
    const int lane = tid & 31;
    const int wave = tid >> 5;
    const int rt   = blockIdx.x >> 2;             // 1024 row tiles
    const int ct   = (blockIdx.x & 3) * 8 + wave; // 32 col strips
    const int row0 = rt * 16;
    const int n00  = ct * 64;
    const int half = lane >> 4;
    const int l15  = lane & 15;

    const v2f* Bi2 = (const v2f*)Bip;

#pragma unroll
    for (int j = 0; j < 2; ++j) {
        int seg = tid + j * 256;
        int r = seg >> 5, c4 = (seg & 31) * 4;
        async_copy_b128(lds_off(&ys[r * YS_STRIDE + c4]),
                        ybuf + (size_t)(row0 + r) * SPECW + c4);
    }
    wait_async_0();
    __syncthreads();

    v8f acc[4] = {};
    const float* yrow = &ys[l15 * YS_STRIDE];

#pragma unroll 2
    for (int kk = 0; kk < SPECW; kk += 4) {
        const int ka = kk + 2 * half;
        const int pa = (kk >> 1) + half;          // pair-row in Bip
        v2f a = *(const v2f*)&yrow[ka];
        v2f bfr[4];
#pragma unroll
        for (int t = 0; t < 4; ++t)
            bfr[t] = Bi2[(size_t)pa * SEQ + n00 + t * 16 + l15];
#pragma unroll
        for (int t = 0; t < 4; ++t)
            acc[t] = wmma_f32(a, bfr[t], acc[t]);
    }
#pragma unroll
    for (int t = 0; t < 4; ++t) {
        const int n0 = n00 + t * 16;
#pragma unroll
        for (int r = 0; r < 8; ++r) {
            const int row = row0 + r + 8 * half;
            out[(size_t)row * SEQ + n0 + l15] = acc[t][r];
        }
    }
}

// ---------------------------------------------------------------------------
extern "C" void kernel_launch(void* const* d_in, const int* in_sizes, int n_in,
                              void* d_out, int out_size, void* d_ws, size_t ws_size,
                              hipStream_t stream) {
    const float* x  = (const float*)d_in[0];   // (32,8,64,2048)
    const float* wr = (const float*)d_in[1];   // (64,64,64)
    const float* wi = (const float*)d_in[2];   // (64,64,64)
    float* out = (float*)d_out;

    // workspace: Bfp | Bip | abuf | ybuf | wrp | wip  (~20 MB)
    float* Bfp  = (float*)d_ws;
    float* Bip  = Bfp  + (size_t)SEQ * SPECW;
    float* abuf = Bip  + (size_t)SPECW * SEQ;
    float* ybuf = abuf + (size_t)MROWS * SPECW;
    float* wrp  = ybuf + (size_t)MROWS * SPECW;
    float* wip  = wrp  + (size_t)NCH * NCH * MODES;

    hipMemsetAsync(abuf, 0, (size_t)MROWS * SPECW * sizeof(float), stream);

    build_basis<<<(SEQ * MODES + 255) / 256, 256, 0, stream>>>(Bfp, Bip);
    repack_w<<<((NCH / 2) * NCH * MODES + 255) / 256, 256, 0, stream>>>(wr, wi, wrp, wip);

    dft_fwd<<<256, 256, 0, stream>>>(x, Bfp, abuf);          // 2048 waves
    mode_mix<<<128, 256, 0, stream>>>(abuf, wrp, wip, ybuf); // 1024 waves
    dft_inv<<<4096, 256, 0, stream>>>(ybuf, Bip, out);       // 32768 waves
}